// verloss_13795434955273
// MI455X (gfx1250) — compile-verified
//
#include <hip/hip_runtime.h>

typedef __attribute__((ext_vector_type(2))) float v2f;
typedef __attribute__((ext_vector_type(8))) float v8f;

namespace {
constexpr int NF   = 16;
constexpr int Hc   = 512, Wc = 640;
constexpr int HPc  = 384, WPc = 512;
constexpr int PADc = 64;
constexpr int HWc  = Hc * Wc;        // 327680
constexpr int PLANE = HPc * WPc;     // 196608 (one y_pred frame)
constexpr int BLOCK = 256;           // 8 wave32s
constexpr int NBLOCKS = HWc / BLOCK; // 1280, exact
}

// Fetch from the implicitly zero-padded HxW view of a (HPc x WPc) plane.
__device__ __forceinline__ float padded_fetch(const float* __restrict__ plane, int yy, int xx) {
  if (yy >= PADc && yy < PADc + HPc && xx >= PADc && xx < PADc + WPc)
    return plane[(yy - PADc) * WPc + (xx - PADc)];
  return 0.0f;
}

__device__ __forceinline__ float bilinear(const float* __restrict__ plane, float gy, float gx) {
  float y = ((gy + 1.0f) * (float)Hc - 1.0f) * 0.5f;
  float x = ((gx + 1.0f) * (float)Wc - 1.0f) * 0.5f;
  float y0f = floorf(y), x0f = floorf(x);
  int y0 = (int)y0f, x0 = (int)x0f;
  float wy1 = y - y0f, wx1 = x - x0f;
  float wy0 = 1.0f - wy1, wx0 = 1.0f - wx1;
  float v00 = padded_fetch(plane, y0, x0);
  float v01 = padded_fetch(plane, y0, x0 + 1);
  float v10 = padded_fetch(plane, y0 + 1, x0);
  float v11 = padded_fetch(plane, y0 + 1, x0 + 1);
  return wy0 * (wx0 * v00 + wx1 * v01) + wy1 * (wx0 * v10 + wx1 * v11);
}

__global__ __launch_bounds__(BLOCK) void verloss_main(
    const float* __restrict__ y_pred, const float* __restrict__ inp1,
    const float* __restrict__ inp2, const float* __restrict__ inp2_back,
    const float* __restrict__ inp7, const float* __restrict__ inp6,
    const float* __restrict__ inp77, const float* __restrict__ Cf,
    const float* __restrict__ Cb, float* __restrict__ partial) {
  const int pix = blockIdx.x * BLOCK + threadIdx.x; // exact cover of H*W
  const int py = pix / Wc, px = pix - py * Wc;

  // Pull this thread's two grid-coordinate cachelines toward the WGP now:
  // all 16 frame coords of C / C_back live in one 128B line each.
  __builtin_prefetch(&Cf[pix * (2 * NF)], 0, 3); // -> global_prefetch_b8
  __builtin_prefetch(&Cb[pix * (2 * NF)], 0, 3);

  const float I0 = inp1[pix];          // inp1[0,0]
  const float I1 = inp1[HWc + pix];    // inp1[0,1]

  const bool inPad = (py >= PADc) && (py < PADc + HPc) && (px >= PADc) && (px < PADc + WPc);
  const int  ppix  = (py - PADc) * WPc + (px - PADc);
  const float padInp6 = inPad ? inp6[ppix] : 0.0f;

  float acc = 0.0f;
  for (int f = 0; f < NF; ++f) {
    const float A2x = inp2[(2 * f + 0) * HWc + pix];
    const float A2y = inp2[(2 * f + 1) * HWc + pix];
    const float B2x = inp2_back[(2 * f + 0) * HWc + pix];
    const float B2y = inp2_back[(2 * f + 1) * HWc + pix];

    // Keep the streaming flow reads one cacheline-trip ahead of demand.
    if (f + 1 < NF) {
      __builtin_prefetch(&inp2[(2 * (f + 1)) * HWc + pix], 0, 1);
      __builtin_prefetch(&inp2_back[(2 * (f + 1)) * HWc + pix], 0, 1);
    }

    float warp5 = 0.0f;
    if (f < NF - 1) { // warp5f[f] = sample(y_pred1[f], grid C[f])
      const v2f cyx = *(const v2f*)&Cf[pix * (2 * NF) + 2 * f]; // one b64 load
      warp5 = bilinear(y_pred + f * PLANE,
                       (cyx.x - (float)(Hc / 2)) * (2.0f / (float)Hc),
                       (cyx.y - (float)(Wc / 2)) * (2.0f / (float)Wc));
    }
    float warp6 = 0.0f;
    if (f >= 1) {     // warp6f[f] = sample(y_pred1[f-1], grid C_back[f])
      const v2f cyx = *(const v2f*)&Cb[pix * (2 * NF) + 2 * f];
      warp6 = bilinear(y_pred + (f - 1) * PLANE,
                       (cyx.x - (float)(Hc / 2)) * (2.0f / (float)Hc),
                       (cyx.y - (float)(Wc / 2)) * (2.0f / (float)Wc));
    }

    const float prev = (f == 0)      ? padInp6
                                     : (inPad ? y_pred[(f - 1) * PLANE + ppix] : 0.0f);
    const float nxt  = (f == NF - 1) ? 0.0f
                                     : (inPad ? y_pred[f * PLANE + ppix] : 0.0f);

    // a[k]=inp5[0][k]+1, b[k]=inp5[3][k], c[k]=inp5[6][k];
    // inp5 = [inp77 col | inp7.reshape(9,15) | zeros col]
    const float af = (f == 0) ? (inp77[0] + 1.0f) : (inp7[f - 1] + 1.0f);
    const float bf = (f == 0) ? inp77[3] : inp7[44 + f];
    const float cf = (f == 0) ? inp77[6] : inp7[89 + f];
    const float an = (f == NF - 1) ? 1.0f : (inp7[f] + 1.0f);
    const float bn = (f == NF - 1) ? 0.0f : inp7[45 + f];
    const float cn = (f == NF - 1) ? 0.0f : inp7[90 + f];

    const float d0 = af * I1 + bf * I0 + cf + prev - (an * A2y + bn * A2x + cn + warp5);
    const float d1 = af * B2y + bf * B2x + cf + warp6 - (an * I1 + bn * I0 + cn + nxt);
    acc = fmaf(d0, d0, acc);
    acc = fmaf(d1, d1, acc);
  }

  // ---- wave32 reduction via one V_WMMA_F32_16X16X4_F32 ----
  // A (16x4): row m = [acc_m, 0, acc_{16+m}, 0];  B (4x16) = ones
  // => D[m][n] = acc_m + acc_{16+m} for every column n.
  v2f A; A.x = acc;  A.y = 0.0f;
  v2f B; B.x = 1.0f; B.y = 1.0f;
  v8f Cacc = {};
  v8f D = __builtin_amdgcn_wmma_f32_16x16x4_f32(false, A, false, B, (short)0, Cacc,
                                                false, false);
  // lanes 0-15 hold rows 0-7, lanes 16-31 hold rows 8-15; fold VGPRs then halves.
  float s = ((D[0] + D[1]) + (D[2] + D[3])) + ((D[4] + D[5]) + (D[6] + D[7]));
  s += __shfl_xor(s, 16, 32); // full wave32 sum in every lane

  __shared__ float wsum[BLOCK / 32];
  const int wave = threadIdx.x >> 5;
  const int lane = threadIdx.x & 31;
  if (lane == 0) wsum[wave] = s;
  __syncthreads();
  if (threadIdx.x == 0) {
    float t = 0.0f;
    for (int i = 0; i < BLOCK / 32; ++i) t += wsum[i];
    partial[blockIdx.x] = t;
  }
}

__global__ __launch_bounds__(BLOCK) void verloss_reduce(const float* __restrict__ partial,
                                                        float* __restrict__ out) {
  __shared__ float sm[BLOCK];
  float s = 0.0f;
  for (int i = threadIdx.x; i < NBLOCKS; i += BLOCK) s += partial[i]; // fixed order
  sm[threadIdx.x] = s;
  __syncthreads();
  for (int off = BLOCK / 2; off > 0; off >>= 1) {
    if ((int)threadIdx.x < off) sm[threadIdx.x] += sm[threadIdx.x + off];
    __syncthreads();
  }
  if (threadIdx.x == 0)
    out[0] = sm[0] * (0.5f / ((float)NF * (float)HWc));
}

extern "C" void kernel_launch(void* const* d_in, const int* in_sizes, int n_in,
                              void* d_out, int out_size, void* d_ws, size_t ws_size,
                              hipStream_t stream) {
  (void)in_sizes; (void)n_in; (void)out_size; (void)ws_size;
  const float* y_pred    = (const float*)d_in[0];
  const float* inp1      = (const float*)d_in[1];
  const float* inp2      = (const float*)d_in[2];
  const float* inp2_back = (const float*)d_in[3];
  const float* inp7      = (const float*)d_in[4];
  const float* inp6      = (const float*)d_in[5];
  const float* inp77     = (const float*)d_in[6];
  const float* C         = (const float*)d_in[9];
  const float* C_back    = (const float*)d_in[10];

  float* partial = (float*)d_ws; // NBLOCKS floats of scratch

  verloss_main<<<NBLOCKS, BLOCK, 0, stream>>>(y_pred, inp1, inp2, inp2_back, inp7,
                                              inp6, inp77, C, C_back, partial);
  verloss_reduce<<<1, BLOCK, 0, stream>>>(partial, (float*)d_out);
}